// WaveletSynthesisHead_67697274519686
// MI455X (gfx1250) — compile-verified
//
#include <hip/hip_runtime.h>
#include <hip/hip_bf16.h>

typedef float v2f __attribute__((ext_vector_type(2)));
typedef float v8f __attribute__((ext_vector_type(8)));

#define B_    16
#define F_    32
#define C_    32
#define T_    2048
#define K_    129
#define RC_   35          // 7*5 (rows x cols table)
#define I_    64          // 2*F channels per group
#define WROWP 160         // padded (reversed) weight row: jr in [-16, 143]
#define ZSTR  260         // padded z row stride (bank-conflict-free b64 reads)
#define NTAU  36          // 144-tau window / K=4 per WMMA
#define NTB   (T_ / 128)  // 16 t-blocks

__global__ __launch_bounds__(256)
void wavelet_synth_wmma(const float* __restrict__ coeffs,
                        const int*   __restrict__ rows,
                        const int*   __restrict__ cols,
                        const float* __restrict__ ri_scale,
                        const float* __restrict__ ri_shift,
                        const float* __restrict__ freq_gain,
                        const float* __restrict__ synth_w,
                        float*       __restrict__ out)
{
    __shared__ float wlds[I_ * WROWP];       // 40,960 B, reversed + zero halo
    __shared__ float zbuf[2][B_ * ZSTR];     // 2 x 16,640 B, transformed z slices

    const int c    = blockIdx.x;             // channel group 0..31
    const int tb   = blockIdx.y;             // 128-wide t block 0..15
    const int T0   = tb * 128;
    const int tid  = threadIdx.x;
    const int wave = tid >> 5;               // 0..7 (wave32)
    const int lane = tid & 31;
    const int half = lane >> 4;
    const int ln   = lane & 15;
    const int bq   = tid >> 4;               // staging: batch row 0..15
    const int tq   = tid & 15;               // staging: 16-tau chunk 0..15

    // ---- weights, stored REVERSED with zero halo:
    //      wlds[i*WROWP + 16 + (128 - j)] = w[c,i,j]
    // so the inner loop reads ascending pairs {jr, jr+1} straight into b.x/b.y.
    for (int idx = tid; idx < I_ * WROWP; idx += 256) wlds[idx] = 0.0f;
    __syncthreads();
    for (int idx = tid; idx < I_ * K_; idx += 256) {
        int i = idx / K_;
        int j = idx - i * K_;
        wlds[i * WROWP + 16 + (128 - j)] = synth_w[(c * I_ + i) * K_ + j];
    }

    const int  r     = rows[c];
    const int  cc    = cols[c];
    const long rcOff = (long)(r * 5 + cc) * T_;
    const bool interior = (tb != 0) && (tb != NTB - 1);   // block-uniform
    const int  tau0  = T0 - 64 + tq * 16;                 // this thread's tau base

    // prefetch registers for the next slice (raw, untransformed)
    float pre[16];

    // issue global loads for slice s (no wait here -> latency hides under WMMAs)
    auto loadSlice = [&](int s) {
        // (bq*2 + ri)*F + f == bq*64 + s  (since s = ri*32 + f)
        const float* grow = coeffs + ((long)(bq * 64 + s)) * ((long)RC_ * T_) + rcOff;
        if (interior) {
            const v2f* src = (const v2f*)(grow + tau0);
            #pragma unroll
            for (int q = 0; q < 8; ++q) {
                v2f v = src[q];
                pre[2 * q]     = v.x;
                pre[2 * q + 1] = v.y;
            }
        } else {
            #pragma unroll
            for (int k = 0; k < 16; ++k) {
                int tau = tau0 + k;
                int tc  = tau < 0 ? 0 : (tau > T_ - 1 ? T_ - 1 : tau);
                pre[k] = grow[tc];                        // clamped, unconditional
            }
        }
    };

    // transform + commit slice s to zbuf[bsel] (consumes pre[])
    auto storeSlice = [&](int s, int bsel) {
        const int   f   = s & 31;
        const float inv = 1.0f / (ri_scale[f] + 1e-12f);
        const float sh  = ri_shift[f];
        const float g   = freq_gain[f];
        float* dst = &zbuf[bsel][bq * ZSTR + tq * 16];
        #pragma unroll
        for (int k = 0; k < 16; ++k) {
            float zv = (pre[k] * inv - sh) * g;
            if (!interior) {
                int tau = tau0 + k;
                if (tau < 0 || tau >= T_) zv = 0.0f;      // exact zero padding
            }
            dst[k] = zv;
        }
    };

    loadSlice(0);
    storeSlice(0, 0);
    __syncthreads();          // weights + slice 0 visible

    v8f acc = (v8f)0.0f;
    const int trel = wave * 16;              // this wave's t-tile offset in block

    for (int s = 0; s < I_; ++s) {
        if (s + 1 < I_) loadSlice(s + 1);    // issue loads, don't wait

        // A: lane ln = batch row; k = 2*half + comp -> taurel = trel + 2*half + 4*ch
        const float* zr = &zbuf[s & 1][ln * ZSTR + trel + 2 * half];
        // B (reversed): jr = 2*half + 4*ch - ln; b.x = wr[jr], b.y = wr[jr+1]
        const float* wr = &wlds[s * WROWP + 16];
        int jr = 2 * half - ln;

        #pragma unroll 6
        for (int ch = 0; ch < NTAU; ++ch) {
            v2f a = *(const v2f*)(zr + 4 * ch);          // ds_load_b64
            v2f b;
            b.x = wr[jr];
            b.y = wr[jr + 1];
            acc = __builtin_amdgcn_wmma_f32_16x16x4_f32(
                      /*neg_a=*/false, a, /*neg_b=*/false, b,
                      /*c_mod=*/(short)0, acc,
                      /*reuse_a=*/false, /*reuse_b=*/false);
            jr += 4;
        }

        if (s + 1 < I_) storeSlice(s + 1, (s + 1) & 1);  // wait lands after WMMAs
        __syncthreads();      // slice s+1 ready; zbuf[s&1] reusable next iter
    }

    // ---- store D: VGPR rr -> batch rr (lanes 0-15) / rr+8 (lanes 16-31) ----
    const int tcol = T0 + trel + ln;
    float* op = out + (long)(half * 8) * C_ * T_ + (long)c * T_ + tcol;
    #pragma unroll
    for (int rr = 0; rr < 8; ++rr) {
        op[(long)rr * C_ * T_] = acc[rr];
    }
}

extern "C" void kernel_launch(void* const* d_in, const int* in_sizes, int n_in,
                              void* d_out, int out_size, void* d_ws, size_t ws_size,
                              hipStream_t stream) {
    const float* coeffs    = (const float*)d_in[0];
    const int*   rows      = (const int*)  d_in[1];
    const int*   cols      = (const int*)  d_in[2];
    const float* ri_scale  = (const float*)d_in[3];
    const float* ri_shift  = (const float*)d_in[4];
    const float* freq_gain = (const float*)d_in[5];
    const float* synth_w   = (const float*)d_in[6];
    float*       out       = (float*)d_out;

    dim3 grid(C_, NTB);
    dim3 block(256);
    wavelet_synth_wmma<<<grid, block, 0, stream>>>(
        coeffs, rows, cols, ri_scale, ri_shift, freq_gain, synth_w, out);
}